// QuantumPerceptron_39084202394272
// MI455X (gfx1250) — compile-verified
//
#include <hip/hip_runtime.h>
#include <hip/hip_bf16.h>
#include <math.h>

typedef __attribute__((ext_vector_type(2))) float v2f;
typedef __attribute__((ext_vector_type(8))) float v8f;

// D = A(16x4) * B(4x16) + C, f32, full precision.
// 8-arg pattern: (neg_a, A, neg_b, B, c_mod, C, reuse_a, reuse_b)
#define WMMA4(a, b, c) \
  __builtin_amdgcn_wmma_f32_16x16x4_f32(false, (a), false, (b), (short)0, (c), false, false)

// Layout: amplitudes (32) x time-columns (16, 9 valid).
// S <- U1 @ S ; S <- H(t) (block-diag, per column) ; S <- U2 @ S, for 4 layers.
__global__ __launch_bounds__(64)
void quantum_perceptron_kernel(const float* __restrict__ state,
                               const float* __restrict__ params,
                               const float* __restrict__ W,
                               const float* __restrict__ bias,
                               float* __restrict__ out)
{
    __shared__ float Sr[32][16];
    __shared__ float Si[32][16];
    __shared__ float Ur[32][32];
    __shared__ float Ui[32][32];
    __shared__ float qre[5][4];
    __shared__ float qim[5][4];
    __shared__ float rvals[16];

    const int tid  = threadIdx.x;
    const int lane = tid & 31;
    const int wave = tid >> 5;
    const int half = lane >> 4;   // 0: lanes 0-15, 1: lanes 16-31
    const int l16  = lane & 15;

    // psi = kron(state, [1,0]) broadcast over the 9 time columns
    for (int idx = tid; idx < 512; idx += 64) {
        int row = idx >> 4, col = idx & 15;
        Sr[row][col] = (((row & 1) == 0) && (col < 9)) ? state[row >> 1] : 0.0f;
        Si[row][col] = 0.0f;
    }
    __syncthreads();

    for (int l = 0; l < 4; ++l) {
        for (int h = 0; h < 2; ++h) {
            // ---- per-qubit 2x2 gates: q = u3(g) u2(b) u1(a) ----
            if (tid < 5) {
                int q = tid;
                float a = params[      l*10 + h*5 + q];
                float b = params[40 +  l*10 + h*5 + q];
                float g = params[80 +  l*10 + h*5 + q];
                float cb = cosf(b),   sb  = sinf(b);
                float cpg = cosf(a+g), spg = sinf(a+g);
                float cmg = cosf(a-g), smg = sinf(a-g);
                qre[q][0] =  cb*cpg;  qim[q][0] = -cb*spg;   // q00 = cb e^{-i(a+g)}
                qre[q][1] =  sb*smg;  qim[q][1] = -sb*cmg;   // q01 = -i sb e^{ i(a-g)}
                qre[q][2] = -sb*smg;  qim[q][2] = -sb*cmg;   // q10 = -i sb e^{-i(a-g)}
                qre[q][3] =  cb*cpg;  qim[q][3] =  cb*spg;   // q11 = cb e^{ i(a+g)}
            }
            __syncthreads();

            // ---- dense U = q0 (x) q1 (x) q2 (x) q3 (x) q4 ----
            for (int idx = tid; idx < 1024; idx += 64) {
                int r = idx >> 5, c = idx & 31;
                float re = 1.0f, im = 0.0f;
                #pragma unroll
                for (int i = 0; i < 5; ++i) {
                    int ri = (r >> (4 - i)) & 1;
                    int ci = (c >> (4 - i)) & 1;
                    float qr = qre[i][ri*2 + ci];
                    float qi = qim[i][ri*2 + ci];
                    float nre = re*qr - im*qi;
                    im = re*qi + im*qr;
                    re = nre;
                }
                Ur[r][c] = re;
                Ui[r][c] = im;
            }
            __syncthreads();

            // ---- S = U @ S (complex) via V_WMMA_F32_16X16X4_F32 ----
            {
                const int m0 = wave << 4;          // row tile per wave
                v8f dr = {}; v8f di = {};
                #pragma unroll
                for (int kb = 0; kb < 32; kb += 4) {
                    int ak = kb + 2*half;          // A/B fragment K index per ISA layout
                    v2f ar  = { Ur[m0 + l16][ak], Ur[m0 + l16][ak + 1] };
                    v2f aip = { Ui[m0 + l16][ak], Ui[m0 + l16][ak + 1] };
                    v2f ain = { -aip.x, -aip.y };  // f32 WMMA has no A-neg: negate in VGPRs
                    v2f br  = { Sr[ak][l16], Sr[ak + 1][l16] };
                    v2f bi  = { Si[ak][l16], Si[ak + 1][l16] };
                    dr = WMMA4(ar,  br, dr);       // + Ur*Sr
                    dr = WMMA4(ain, bi, dr);       // - Ui*Si
                    di = WMMA4(ar,  bi, di);       // + Ur*Si
                    di = WMMA4(aip, br, di);       // + Ui*Sr
                }
                __syncthreads();                   // all reads of S complete (both waves)
                #pragma unroll
                for (int v = 0; v < 8; ++v) {      // C/D layout: row = v + 8*half
                    int row = m0 + v + 8*half;
                    Sr[row][l16] = dr[v];
                    Si[row][l16] = di[v];
                }
                __syncthreads();
            }

            // ---- after U1: apply H(t) = expm(-i t M), block-diag 2x2 ----
            if (h == 0) {
                for (int idx = tid; idx < 256; idx += 64) {
                    int k = idx >> 4;              // pair (2k, 2k+1)
                    int t = idx & 15;              // time column
                    float time  = 0.1f * (float)(t + 1);
                    float delta = 4.0f - 2.0f * (float)__popc(k);
                    float nu = sqrtf(delta*delta + 400.0f);
                    float ct = cosf(time*nu);
                    float sv = sinf(time*nu) / nu;
                    float pr = cosf(4.0f*time);    // e^{-i t mu}, mu = -4
                    float pi = sinf(4.0f*time);
                    // pre-phase block: b00 = ct - i*delta*sv ; b01 = -i*(-20)*sv ; b11 = ct + i*delta*sv
                    float b00i = -delta*sv, b01i = 20.0f*sv, b11i = delta*sv;
                    float B00r = ct*pr - b00i*pi,   B00i = ct*pi + b00i*pr;
                    float B01r =       - b01i*pi,   B01i =         b01i*pr;
                    float B11r = ct*pr - b11i*pi,   B11i = ct*pi + b11i*pr;
                    float x0r = Sr[2*k][t],   x0i = Si[2*k][t];
                    float x1r = Sr[2*k+1][t], x1i = Si[2*k+1][t];
                    float y0r = B00r*x0r - B00i*x0i + B01r*x1r - B01i*x1i;
                    float y0i = B00r*x0i + B00i*x0r + B01r*x1i + B01i*x1r;
                    float y1r = B01r*x0r - B01i*x0i + B11r*x1r - B11i*x1i;   // B10 == B01
                    float y1i = B01r*x0i + B01i*x0r + B11r*x1i + B11i*x1r;
                    Sr[2*k][t]   = y0r;  Si[2*k][t]   = y0i;
                    Sr[2*k+1][t] = y1r;  Si[2*k+1][t] = y1i;
                }
                __syncthreads();
            }
        }
    }

    // ---- readout: r[t] = sum_even |psi|^2 - sum_odd |psi|^2 ----
    if (tid < 16) {
        float acc = 0.0f;
        for (int row = 0; row < 32; ++row) {
            float p = Sr[row][tid]*Sr[row][tid] + Si[row][tid]*Si[row][tid];
            acc += (row & 1) ? -p : p;
        }
        rvals[tid] = acc;
    }
    __syncthreads();
    if (tid < 2) {
        float acc = bias[tid];
        for (int t = 0; t < 9; ++t) acc += rvals[t] * W[tid*9 + t];
        out[tid] = tanhf(acc);
    }
}

extern "C" void kernel_launch(void* const* d_in, const int* in_sizes, int n_in,
                              void* d_out, int out_size, void* d_ws, size_t ws_size,
                              hipStream_t stream) {
    const float* state  = (const float*)d_in[0];   // (16,)  f32
    const float* params = (const float*)d_in[1];   // (3,4,2,5) f32
    const float* W      = (const float*)d_in[2];   // (2,9)  f32
    const float* bias   = (const float*)d_in[3];   // (2,)   f32
    float* out = (float*)d_out;                    // (1,2)  f32

    quantum_perceptron_kernel<<<dim3(1), dim3(64), 0, stream>>>(state, params, W, bias, out);
}